// PR3d_35888746726112
// MI455X (gfx1250) — compile-verified
//
#include <hip/hip_runtime.h>

#define NTT 20
#define SDIM 14
#define MMODE 4
#define BATCH 32
#define CINC 16
#define COUTC 16
#define NF (NTT*SDIM*SDIM)     // 3920
#define NCH (BATCH*CINC)       // 512
#define TWO_PI 6.28318530717958647692f

typedef float v2f __attribute__((ext_vector_type(2)));
typedef float v8f __attribute__((ext_vector_type(8)));

__device__ __forceinline__ void cmac(float& dr, float& di, float ar, float ai, float br, float bi) {
    dr = fmaf(ar, br, fmaf(-ai, bi, dr));
    di = fmaf(ar, bi, fmaf(ai, br, di));
}

// ---------------------------------------------------------------------------
// Kernel 1: precompute invA[o][i][k][p], invB[x][i][k][q], invC[y][i][k][r]
// (complex reciprocals of lam - pole) and e1[i][k][p][z], e2[i][k][q][x],
// e3[i][k][r][y] = exp(pole * grid).
// ---------------------------------------------------------------------------
__global__ __launch_bounds__(256) void prep_kernel(
    const float* __restrict__ p1, const float* __restrict__ p2, const float* __restrict__ p3,
    float* invARe, float* invAIm, float* invBRe, float* invBIm, float* invCRe, float* invCIm,
    float* e1Re, float* e1Im, float* e2Re, float* e2Im, float* e3Re, float* e3Im) {
    int tid = threadIdx.x;
    // invA: 20*16*16*4 = 20480   lam1 = 2*pi*i*fftfreq(20, d=1)
    for (int idx = tid; idx < NTT*CINC*COUTC*MMODE; idx += blockDim.x) {
        int p = idx & 3, k = (idx >> 2) & 15, i = (idx >> 6) & 15, o = idx >> 10;
        float fo = (o < NTT/2) ? (float)o : (float)(o - NTT);
        float lam = TWO_PI * fo / (float)NTT;
        int w = (i*COUTC + k)*MMODE + p;
        float dr = -p1[2*w], di = lam - p1[2*w+1];
        float s = 1.0f / (dr*dr + di*di);
        invARe[idx] = dr*s; invAIm[idx] = -di*s;
    }
    // invB / invC: 14*16*16*4 = 14336   lam = 2*pi*i*fftfreq(14, d=1/27)
    for (int idx = tid; idx < SDIM*CINC*COUTC*MMODE; idx += blockDim.x) {
        int p = idx & 3, k = (idx >> 2) & 15, i = (idx >> 6) & 15, x = idx >> 10;
        float fx = ((x < SDIM/2) ? (float)x : (float)(x - SDIM)) * (27.0f / (float)SDIM);
        float lam = TWO_PI * fx;
        int w = (i*COUTC + k)*MMODE + p;
        float dr2 = -p2[2*w], di2 = lam - p2[2*w+1];
        float s2 = 1.0f / (dr2*dr2 + di2*di2);
        invBRe[idx] = dr2*s2; invBIm[idx] = -di2*s2;
        float dr3 = -p3[2*w], di3 = lam - p3[2*w+1];
        float s3 = 1.0f / (dr3*dr3 + di3*di3);
        invCRe[idx] = dr3*s3; invCIm[idx] = -di3*s3;
    }
    // e1: [i][k][p][z], t = z
    for (int idx = tid; idx < CINC*COUTC*MMODE*NTT; idx += blockDim.x) {
        int z = idx % NTT, m = idx / NTT;
        float t = (float)z;
        float pr = p1[2*m], pi = p1[2*m+1];
        float mag = expf(pr * t);
        e1Re[idx] = mag * cosf(pi * t); e1Im[idx] = mag * sinf(pi * t);
    }
    // e2/e3: [i][k][q][x], t = x/27
    for (int idx = tid; idx < CINC*COUTC*MMODE*SDIM; idx += blockDim.x) {
        int x = idx % SDIM, m = idx / SDIM;
        float t = (float)x * (1.0f/27.0f);
        float pr2 = p2[2*m], pi2 = p2[2*m+1];
        float mg2 = expf(pr2 * t);
        e2Re[idx] = mg2 * cosf(pi2 * t); e2Im[idx] = mg2 * sinf(pi2 * t);
        float pr3 = p3[2*m], pi3 = p3[2*m+1];
        float mg3 = expf(pr3 * t);
        e3Re[idx] = mg3 * cosf(pi3 * t); e3Im[idx] = mg3 * sinf(pi3 * t);
    }
}

// ---------------------------------------------------------------------------
// Kernel 2: forward 3D DFT of x, one (batch,cin) slice per block, three
// matmul passes through LDS. Output alpha planar, frequency-major: [f][b*16+i]
// ---------------------------------------------------------------------------
__global__ __launch_bounds__(256) void dft_fwd_kernel(const float* __restrict__ x,
                                                      float* __restrict__ aRe,
                                                      float* __restrict__ aIm) {
    __shared__ float sR[NF], sI[NF], dR[NF], dI[NF];
    __shared__ float c20[NTT], s20[NTT], c14[SDIM], s14[SDIM];
    int bc = blockIdx.x, tid = threadIdx.x;
    if (tid < NTT) { float th = TWO_PI * tid / (float)NTT; c20[tid] = cosf(th); s20[tid] = sinf(th); }
    if (tid >= 32 && tid < 32 + SDIM) {
        int j = tid - 32; float th = TWO_PI * j / (float)SDIM; c14[j] = cosf(th); s14[j] = sinf(th);
    }
    for (int n = tid; n < NF; n += blockDim.x) { sR[n] = x[bc*NF + n]; sI[n] = 0.0f; }
    __syncthreads();
    // t axis (forward: W = cos - i sin)
    for (int idx = tid; idx < NF; idx += blockDim.x) {
        int col = idx % 196, o = idx / 196;
        float dr = 0.f, di = 0.f;
        for (int t = 0; t < NTT; ++t) {
            int j = (o * t) % NTT; float c = c20[j], s = s20[j];
            float ar = sR[t*196 + col], ai = sI[t*196 + col];
            dr += ar*c + ai*s; di += ai*c - ar*s;
        }
        dR[idx] = dr; dI[idx] = di;
    }
    __syncthreads();
    // x axis
    for (int idx = tid; idx < NF; idx += blockDim.x) {
        int y = idx % SDIM, xo = (idx / SDIM) % SDIM, o = idx / 196;
        float dr = 0.f, di = 0.f;
        for (int xx = 0; xx < SDIM; ++xx) {
            int j = (xo * xx) % SDIM; float c = c14[j], s = s14[j];
            float ar = dR[o*196 + xx*SDIM + y], ai = dI[o*196 + xx*SDIM + y];
            dr += ar*c + ai*s; di += ai*c - ar*s;
        }
        sR[idx] = dr; sI[idx] = di;
    }
    __syncthreads();
    // y axis + store frequency-major
    for (int idx = tid; idx < NF; idx += blockDim.x) {
        int yo = idx % SDIM, row = idx / SDIM;
        float dr = 0.f, di = 0.f;
        for (int yy = 0; yy < SDIM; ++yy) {
            int j = (yo * yy) % SDIM; float c = c14[j], s = s14[j];
            float ar = sR[row*SDIM + yy], ai = sI[row*SDIM + yy];
            dr += ar*c + ai*s; di += ai*c - ar*s;
        }
        aRe[idx*NCH + bc] = dr; aIm[idx*NCH + bc] = di;
    }
}

// ---------------------------------------------------------------------------
// Kernel 3: G[f][i*16+k] = sum_pqr residue[i,k,p,q,r]*invA[o,p]*invB[x,q]*invC[y,r]
// via separable contraction, one (i,k) per block.
// ---------------------------------------------------------------------------
__global__ __launch_bounds__(128) void g_kernel(const float* __restrict__ res,
    const float* __restrict__ invARe, const float* __restrict__ invAIm,
    const float* __restrict__ invBRe, const float* __restrict__ invBIm,
    const float* __restrict__ invCRe, const float* __restrict__ invCIm,
    float* __restrict__ gRe, float* __restrict__ gIm) {
    __shared__ float rR[64], rI[64];
    __shared__ float aR[NTT*MMODE], aI[NTT*MMODE];
    __shared__ float bR[SDIM*MMODE], bI[SDIM*MMODE];
    __shared__ float cR[SDIM*MMODE], cI[SDIM*MMODE];
    __shared__ float t1R[16*SDIM], t1I[16*SDIM];      // [p*4+q][s]
    __shared__ float t2R[MMODE*196], t2I[MMODE*196];  // [p][x*14+s]
    int i = blockIdx.x >> 4, k = blockIdx.x & 15, tid = threadIdx.x;
    for (int t = tid; t < 64; t += blockDim.x) {
        int w = (i*16 + k)*64 + t; rR[t] = res[2*w]; rI[t] = res[2*w+1];
    }
    for (int t = tid; t < NTT*MMODE; t += blockDim.x) {
        int o = t >> 2, p = t & 3; int idx = (o*256 + i*16 + k)*4 + p;
        aR[t] = invARe[idx]; aI[t] = invAIm[idx];
    }
    for (int t = tid; t < SDIM*MMODE; t += blockDim.x) {
        int x = t >> 2, q = t & 3; int idx = (x*256 + i*16 + k)*4 + q;
        bR[t] = invBRe[idx]; bI[t] = invBIm[idx];
        cR[t] = invCRe[idx]; cI[t] = invCIm[idx];
    }
    __syncthreads();
    for (int t = tid; t < 16*SDIM; t += blockDim.x) {   // sum over r
        int s = t % SDIM, pq = t / SDIM;
        float dr = 0.f, di = 0.f;
        for (int r = 0; r < 4; ++r) cmac(dr, di, rR[pq*4+r], rI[pq*4+r], cR[s*4+r], cI[s*4+r]);
        t1R[t] = dr; t1I[t] = di;
    }
    __syncthreads();
    for (int t = tid; t < MMODE*196; t += blockDim.x) { // sum over q
        int s = t % SDIM, x = (t / SDIM) % SDIM, p = t / 196;
        float dr = 0.f, di = 0.f;
        for (int q = 0; q < 4; ++q) cmac(dr, di, t1R[(p*4+q)*SDIM+s], t1I[(p*4+q)*SDIM+s], bR[x*4+q], bI[x*4+q]);
        t2R[t] = dr; t2I[t] = di;
    }
    __syncthreads();
    for (int f = tid; f < NF; f += blockDim.x) {        // sum over p
        int xs = f % 196, o = f / 196;
        float dr = 0.f, di = 0.f;
        for (int p = 0; p < 4; ++p) cmac(dr, di, t2R[p*196+xs], t2I[p*196+xs], aR[o*4+p], aI[o*4+p]);
        gRe[f*256 + i*16 + k] = dr; gIm[f*256 + i*16 + k] = di;
    }
}

// ---------------------------------------------------------------------------
// Kernel 4: per-frequency complex matmul out_res1 = alpha @ G via WMMA f32
// 16x16x4. One wave per (frequency, 16-row tile of the 32-row batch dim).
// 7840 waves total; all lanes active (EXEC all-ones as WMMA requires).
// ---------------------------------------------------------------------------
__global__ __launch_bounds__(128) void wmma_res1_kernel(
    const float* __restrict__ aRe, const float* __restrict__ aIm,
    const float* __restrict__ gRe, const float* __restrict__ gIm,
    float* __restrict__ oRe, float* __restrict__ oIm) {
    int wave = (blockIdx.x * blockDim.x + threadIdx.x) >> 5;
    int lane = threadIdx.x & 31;
    int f = wave >> 1;
    int mt = wave & 1;                 // row tile: batches [mt*16, mt*16+16)
    int r = lane & 15, hi = lane >> 4;
    const float* aReF = aRe + (size_t)f * NCH;
    const float* aImF = aIm + (size_t)f * NCH;
    const float* gReF = gRe + (size_t)f * 256;
    const float* gImF = gIm + (size_t)f * 256;
    v2f Ar[4], Ai[4], AiN[4], Br[4], Bi[4];
#pragma unroll
    for (int c = 0; c < 4; ++c) {
        // A(16x4) chunk: lane holds A[M=r, K=4c+2*hi .. +1]
        int arow = (mt*16 + r)*16 + 4*c + 2*hi;
        Ar[c][0] = aReF[arow];   Ar[c][1] = aReF[arow + 1];
        Ai[c][0] = aImF[arow];   Ai[c][1] = aImF[arow + 1];
        AiN[c] = -Ai[c];         // f32 WMMA NEG only supports CNeg -> negate A in VALU
        // B(4x16) chunk: lane holds B[K=4c+2*hi][N=r] and B[K+1][N=r]
        int bk = (4*c + 2*hi)*16 + r;
        Br[c][0] = gReF[bk];     Br[c][1] = gReF[bk + 16];
        Bi[c][0] = gImF[bk];     Bi[c][1] = gImF[bk + 16];
    }
    v8f cre = {0,0,0,0,0,0,0,0};
    v8f cim = {0,0,0,0,0,0,0,0};
#pragma unroll
    for (int c = 0; c < 4; ++c)
        cre = __builtin_amdgcn_wmma_f32_16x16x4_f32(false, Ar[c], false, Br[c], (short)0, cre, false, false);
#pragma unroll
    for (int c = 0; c < 4; ++c)
        cre = __builtin_amdgcn_wmma_f32_16x16x4_f32(false, AiN[c], false, Bi[c], (short)0, cre, false, false);
#pragma unroll
    for (int c = 0; c < 4; ++c)
        cim = __builtin_amdgcn_wmma_f32_16x16x4_f32(false, Ar[c], false, Bi[c], (short)0, cim, false, false);
#pragma unroll
    for (int c = 0; c < 4; ++c)
        cim = __builtin_amdgcn_wmma_f32_16x16x4_f32(false, Ai[c], false, Br[c], (short)0, cim, false, false);
    // D(16x16): lane<16 -> (M=j, N=lane); lane>=16 -> (M=j+8, N=lane-16)
#pragma unroll
    for (int j = 0; j < 8; ++j) {
        int row = mt*16 + j + 8*hi;
        size_t addr = (size_t)f * NCH + row*16 + r;
        oRe[addr] = cre[j]; oIm[addr] = cim[j];
    }
}

// ---------------------------------------------------------------------------
// Kernel 5: out_res2[b,k,pqr] = -sum_i residue[i,k,pqr] * <alpha[b,i] | pole basis>
// separable projection, one (batch b, cout k) per block, loop over cin i.
// ---------------------------------------------------------------------------
__global__ __launch_bounds__(128) void proj_kernel(
    const float* __restrict__ aRe, const float* __restrict__ aIm,
    const float* __restrict__ invARe, const float* __restrict__ invAIm,
    const float* __restrict__ invBRe, const float* __restrict__ invBIm,
    const float* __restrict__ invCRe, const float* __restrict__ invCIm,
    const float* __restrict__ res, float* __restrict__ o2Re, float* __restrict__ o2Im) {
    __shared__ float xR[NF], xI[NF];
    __shared__ float t1R[1120], t1I[1120];  // [(o*14+x)*4+r]
    __shared__ float t2R[320], t2I[320];    // [(o*4+q)*4+r]
    __shared__ float accR[64], accI[64];
    __shared__ float aR[80], aI[80], bR[56], bI[56], cR[56], cI[56];
    int b = blockIdx.x >> 4, k = blockIdx.x & 15, tid = threadIdx.x;
    for (int t = tid; t < 64; t += blockDim.x) { accR[t] = 0.f; accI[t] = 0.f; }
    for (int i = 0; i < CINC; ++i) {
        __syncthreads();
        for (int n = tid; n < NF; n += blockDim.x) {
            size_t a = (size_t)n * NCH + b*16 + i; xR[n] = aRe[a]; xI[n] = aIm[a];
        }
        for (int t = tid; t < 80; t += blockDim.x) {
            int o = t >> 2, p = t & 3; int idx = (o*256 + i*16 + k)*4 + p;
            aR[t] = invARe[idx]; aI[t] = invAIm[idx];
        }
        for (int t = tid; t < 56; t += blockDim.x) {
            int xx = t >> 2, q = t & 3; int idx = (xx*256 + i*16 + k)*4 + q;
            bR[t] = invBRe[idx]; bI[t] = invBIm[idx];
            cR[t] = invCRe[idx]; cI[t] = invCIm[idx];
        }
        __syncthreads();
        for (int t = tid; t < 1120; t += blockDim.x) {  // contract y with invC
            int r = t & 3, ox = t >> 2;
            float dr = 0.f, di = 0.f;
            for (int s = 0; s < SDIM; ++s) cmac(dr, di, xR[ox*SDIM+s], xI[ox*SDIM+s], cR[s*4+r], cI[s*4+r]);
            t1R[t] = dr; t1I[t] = di;
        }
        __syncthreads();
        for (int t = tid; t < 320; t += blockDim.x) {   // contract x with invB
            int r = t & 3, q = (t >> 2) & 3, o = t >> 4;
            float dr = 0.f, di = 0.f;
            for (int xx = 0; xx < SDIM; ++xx)
                cmac(dr, di, t1R[(o*SDIM+xx)*4+r], t1I[(o*SDIM+xx)*4+r], bR[xx*4+q], bI[xx*4+q]);
            t2R[t] = dr; t2I[t] = di;
        }
        __syncthreads();
        for (int t = tid; t < 64; t += blockDim.x) {    // contract t with invA, fold residue
            int r = t & 3, q = (t >> 2) & 3, p = t >> 4;
            float dr = 0.f, di = 0.f;
            for (int o = 0; o < NTT; ++o)
                cmac(dr, di, t2R[(o*4+q)*4+r], t2I[(o*4+q)*4+r], aR[o*4+p], aI[o*4+p]);
            int w = (i*16 + k)*64 + t;
            float rr = res[2*w], ri = res[2*w+1];
            accR[t] += rr*dr - ri*di;
            accI[t] += rr*di + ri*dr;
        }
    }
    __syncthreads();
    for (int t = tid; t < 64; t += blockDim.x) {
        o2Re[(b*16 + k)*64 + t] = -accR[t];
        o2Im[(b*16 + k)*64 + t] = -accI[t];
    }
}

// ---------------------------------------------------------------------------
// Kernel 6: x2[kb, io, z, x, y] = sum_b sum_pqm out_res2[kb,b,pqm]*e1*e2*e3  (real)
// separable Tucker reconstruction, one (batch kb, cout io) per block.
// ---------------------------------------------------------------------------
__global__ __launch_bounds__(128) void x2_kernel(
    const float* __restrict__ o2Re, const float* __restrict__ o2Im,
    const float* __restrict__ e1Re, const float* __restrict__ e1Im,
    const float* __restrict__ e2Re, const float* __restrict__ e2Im,
    const float* __restrict__ e3Re, const float* __restrict__ e3Im,
    float* __restrict__ x2buf) {
    __shared__ float accR[NF];
    __shared__ float coR[64], coI[64];
    __shared__ float v1R[320], v1I[320];    // [(q*4+m)*20+z]
    __shared__ float v2R[1120], v2I[1120];  // [(m*20+z)*14+x]
    __shared__ float E1R[80], E1I[80], E2R[56], E2I[56], E3R[56], E3I[56];
    int kb = blockIdx.x >> 4, io = blockIdx.x & 15, tid = threadIdx.x;
    for (int n = tid; n < NF; n += blockDim.x) accR[n] = 0.f;
    for (int b = 0; b < 16; ++b) {
        __syncthreads();
        for (int t = tid; t < 64; t += blockDim.x) {
            int w = (kb*16 + b)*64 + t; coR[t] = o2Re[w]; coI[t] = o2Im[w];
        }
        for (int t = tid; t < 80; t += blockDim.x) {
            int p = t / NTT, z = t % NTT; int idx = ((b*16 + io)*4 + p)*NTT + z;
            E1R[p*NTT+z] = e1Re[idx]; E1I[p*NTT+z] = e1Im[idx];
        }
        for (int t = tid; t < 56; t += blockDim.x) {
            int q = t / SDIM, xx = t % SDIM; int idx = ((b*16 + io)*4 + q)*SDIM + xx;
            E2R[t] = e2Re[idx]; E2I[t] = e2Im[idx];
            E3R[t] = e3Re[idx]; E3I[t] = e3Im[idx];
        }
        __syncthreads();
        for (int t = tid; t < 320; t += blockDim.x) {   // contract p with e1
            int z = t % NTT; int qm = t / NTT; int m = qm & 3, q = qm >> 2;
            float dr = 0.f, di = 0.f;
            for (int p = 0; p < 4; ++p)
                cmac(dr, di, coR[(p*4+q)*4+m], coI[(p*4+q)*4+m], E1R[p*NTT+z], E1I[p*NTT+z]);
            v1R[t] = dr; v1I[t] = di;
        }
        __syncthreads();
        for (int t = tid; t < 1120; t += blockDim.x) {  // contract q with e2
            int xx = t % SDIM; int mz = t / SDIM; int z = mz % NTT; int m = mz / NTT;
            float dr = 0.f, di = 0.f;
            for (int q = 0; q < 4; ++q)
                cmac(dr, di, v1R[(q*4+m)*NTT+z], v1I[(q*4+m)*NTT+z], E2R[q*SDIM+xx], E2I[q*SDIM+xx]);
            v2R[t] = dr; v2I[t] = di;
        }
        __syncthreads();
        for (int t = tid; t < NF; t += blockDim.x) {    // contract m with e3, keep real
            int y = t % SDIM; int zx = t / SDIM; int xx = zx % SDIM; int z = zx / SDIM;
            float dr = 0.f;
            for (int m = 0; m < 4; ++m)
                dr += v2R[(m*NTT+z)*SDIM+xx]*E3R[m*SDIM+y] - v2I[(m*NTT+z)*SDIM+xx]*E3I[m*SDIM+y];
            accR[t] += dr;
        }
    }
    __syncthreads();
    for (int n = tid; n < NF; n += blockDim.x) x2buf[(size_t)(kb*16 + io)*NF + n] = accR[n];
}

// ---------------------------------------------------------------------------
// Kernel 7: inverse 3D DFT of out_res1 (one (batch,cout) slice per block),
// add x2, scale by 1/3920, write final float32 output.
// ---------------------------------------------------------------------------
__global__ __launch_bounds__(256) void idft_combine_kernel(
    const float* __restrict__ specRe, const float* __restrict__ specIm,
    const float* __restrict__ x2buf, float* __restrict__ out) {
    __shared__ float sR[NF], sI[NF], dR[NF], dI[NF];
    __shared__ float c20[NTT], s20[NTT], c14[SDIM], s14[SDIM];
    int bc = blockIdx.x, tid = threadIdx.x;
    if (tid < NTT) { float th = TWO_PI * tid / (float)NTT; c20[tid] = cosf(th); s20[tid] = sinf(th); }
    if (tid >= 32 && tid < 32 + SDIM) {
        int j = tid - 32; float th = TWO_PI * j / (float)SDIM; c14[j] = cosf(th); s14[j] = sinf(th);
    }
    for (int n = tid; n < NF; n += blockDim.x) {
        sR[n] = specRe[(size_t)n*NCH + bc]; sI[n] = specIm[(size_t)n*NCH + bc];
    }
    __syncthreads();
    // t axis (inverse: W = cos + i sin)
    for (int idx = tid; idx < NF; idx += blockDim.x) {
        int col = idx % 196, o = idx / 196;
        float dr = 0.f, di = 0.f;
        for (int t = 0; t < NTT; ++t) {
            int j = (o * t) % NTT; float c = c20[j], s = s20[j];
            float ar = sR[t*196 + col], ai = sI[t*196 + col];
            dr += ar*c - ai*s; di += ai*c + ar*s;
        }
        dR[idx] = dr; dI[idx] = di;
    }
    __syncthreads();
    for (int idx = tid; idx < NF; idx += blockDim.x) {
        int y = idx % SDIM, xo = (idx / SDIM) % SDIM, o = idx / 196;
        float dr = 0.f, di = 0.f;
        for (int xx = 0; xx < SDIM; ++xx) {
            int j = (xo * xx) % SDIM; float c = c14[j], s = s14[j];
            float ar = dR[o*196 + xx*SDIM + y], ai = dI[o*196 + xx*SDIM + y];
            dr += ar*c - ai*s; di += ai*c + ar*s;
        }
        sR[idx] = dr; sI[idx] = di;
    }
    __syncthreads();
    const float invN = 1.0f / (float)NF;
    for (int idx = tid; idx < NF; idx += blockDim.x) {
        int yo = idx % SDIM, row = idx / SDIM;
        float dr = 0.f;
        for (int yy = 0; yy < SDIM; ++yy) {
            int j = (yo * yy) % SDIM; float c = c14[j], s = s14[j];
            dr += sR[row*SDIM + yy]*c - sI[row*SDIM + yy]*s;
        }
        out[(size_t)bc*NF + idx] = (dr + x2buf[(size_t)bc*NF + idx]) * invN;
    }
}

// ---------------------------------------------------------------------------
extern "C" void kernel_launch(void* const* d_in, const int* in_sizes, int n_in,
                              void* d_out, int out_size, void* d_ws, size_t ws_size,
                              hipStream_t stream) {
    (void)in_sizes; (void)n_in; (void)out_size; (void)ws_size;
    const float* x   = (const float*)d_in[0];
    const float* p1  = (const float*)d_in[1];  // complex64 interleaved
    const float* p2  = (const float*)d_in[2];
    const float* p3  = (const float*)d_in[3];
    const float* res = (const float*)d_in[4];
    float* out = (float*)d_out;

    float* w = (float*)d_ws;
    size_t off = 0;
    auto alloc = [&](size_t n) { float* p = w + off; off += (n + 63) & ~(size_t)63; return p; };
    float* alphaRe = alloc((size_t)NF*NCH);
    float* alphaIm = alloc((size_t)NF*NCH);
    float* gRe     = alloc((size_t)NF*256);
    float* gIm     = alloc((size_t)NF*256);
    float* specRe  = alloc((size_t)NF*NCH);
    float* specIm  = alloc((size_t)NF*NCH);
    float* o2Re    = alloc(32*16*64);
    float* o2Im    = alloc(32*16*64);
    float* x2buf   = alloc((size_t)NCH*NF);
    float* invARe  = alloc(NTT*256*4);  float* invAIm = alloc(NTT*256*4);
    float* invBRe  = alloc(SDIM*256*4); float* invBIm = alloc(SDIM*256*4);
    float* invCRe  = alloc(SDIM*256*4); float* invCIm = alloc(SDIM*256*4);
    float* e1Re    = alloc(256*4*NTT);  float* e1Im   = alloc(256*4*NTT);
    float* e2Re    = alloc(256*4*SDIM); float* e2Im   = alloc(256*4*SDIM);
    float* e3Re    = alloc(256*4*SDIM); float* e3Im   = alloc(256*4*SDIM);

    prep_kernel<<<1, 256, 0, stream>>>(p1, p2, p3,
        invARe, invAIm, invBRe, invBIm, invCRe, invCIm,
        e1Re, e1Im, e2Re, e2Im, e3Re, e3Im);
    dft_fwd_kernel<<<NCH, 256, 0, stream>>>(x, alphaRe, alphaIm);
    g_kernel<<<256, 128, 0, stream>>>(res, invARe, invAIm, invBRe, invBIm, invCRe, invCIm, gRe, gIm);
    wmma_res1_kernel<<<(NF*2*32)/128, 128, 0, stream>>>(alphaRe, alphaIm, gRe, gIm, specRe, specIm);
    proj_kernel<<<BATCH*COUTC, 128, 0, stream>>>(alphaRe, alphaIm,
        invARe, invAIm, invBRe, invBIm, invCRe, invCIm, res, o2Re, o2Im);
    x2_kernel<<<BATCH*COUTC, 128, 0, stream>>>(o2Re, o2Im,
        e1Re, e1Im, e2Re, e2Im, e3Re, e3Im, x2buf);
    idft_combine_kernel<<<NCH, 256, 0, stream>>>(specRe, specIm, x2buf, out);
}